// TopNGating_64536178590139
// MI455X (gfx1250) — compile-verified
//
#include <hip/hip_runtime.h>

// ---------------------------------------------------------------------------
// TopN (top-2) MoE gating for MI455X / gfx1250.
//   B=2, N=4096, D=2048, E=8, K=2, CAP = int(4096*1.25/8) = 640
//   d_out = [dispatch (2*4096*8*640) | combine (same) | balance_loss | z_loss]
// Strategy (roofline): 335MB output zero-fill dominates (~14us @23.3TB/s);
// GEMM is 268 MFLOP -> do it with V_WMMA_F32_16X16X4_F32 fused with softmax,
// then LDS-scan + sparse scatter of the ~16K nonzero entries.
// ---------------------------------------------------------------------------

typedef __attribute__((ext_vector_type(2))) float v2f;
typedef __attribute__((ext_vector_type(4))) float v4f;
typedef __attribute__((ext_vector_type(8))) float v8f;

#define NB      2
#define NN      4096
#define NDIM    2048
#define NE      8
#define NTOK    (NB * NN)            // 8192
#define CAP     640
#define NDC     41943040L            // NB*NN*NE*CAP (size of dispatch == combine)
#define OUT_F4  20971520UL           // (2*NDC)/4 float4s to zero
// workspace layout (floats): [0..32767] token records (v4f per token),
// [32768..32783] rawsum[b][e], [32784] zsum, [32800..32815] cnt1[b][e]
#define WS_ACC_OFF 32768

// ---------------- Kernel 0: zero-fill output + workspace accumulators -------
__global__ void k_zero(float* __restrict__ out, float* __restrict__ wsacc) {
    const v4f z = {0.f, 0.f, 0.f, 0.f};
    v4f* o4 = (v4f*)out;
    size_t tid = (size_t)blockIdx.x * blockDim.x + threadIdx.x;
    size_t stride = (size_t)gridDim.x * blockDim.x;
    for (size_t i = tid; i < OUT_F4; i += stride)
        __builtin_nontemporal_store(z, &o4[i]);
    if (tid < 2)  out[2L * NDC + tid] = 0.f;   // scalar loss slots
    if (tid < 64) wsacc[tid] = 0.f;            // rawsum/zsum/cnt1 accumulators
}

// ---------------- Kernel 1: WMMA gemm + softmax + top-2 + routing records ---
// One wave per 16-token tile; 512 waves total (64 blocks x 8 waves).
__global__ void k_gate(const float* __restrict__ x, const float* __restrict__ W,
                       const float* __restrict__ probs, v4f* __restrict__ rec,
                       float* __restrict__ rawsum, float* __restrict__ zsum) {
    __shared__ float lgt[8][16][17];           // [wave-in-block][token][expert] (+pad)

    const int lane = threadIdx.x & 31;
    const int wib  = threadIdx.x >> 5;
    const int wave = blockIdx.x * 8 + wib;
    const int tokenBase = wave << 4;           // 16 tokens per wave
    const int m = lane & 15;
    const int h = lane >> 4;

    // A operand: lane (m + 16h) holds x[tile_row m][k0 + 2h + {0,1}]
    const float* xr = x + (size_t)(tokenBase + m) * NDIM + 2 * h;
    // B operand: lane (n + 16h) holds W^T[k0+2h+{0,1}][n] = W[n][k0+2h+{0,1}]
    // pad experts 8..15 with zero via multiply (keeps EXEC all-ones at WMMA)
    const float* wr = W + (size_t)(m & 7) * NDIM + 2 * h;
    const float bscale = (m < 8) ? 1.f : 0.f;

    v8f acc = {0.f, 0.f, 0.f, 0.f, 0.f, 0.f, 0.f, 0.f};
#pragma unroll 4
    for (int k0 = 0; k0 < NDIM; k0 += 4) {
        v2f a  = *(const v2f*)(xr + k0);
        v2f bb = *(const v2f*)(wr + k0);
        bb.x *= bscale; bb.y *= bscale;
        // D = A(16x4,f32) * B(4x16,f32) + C(16x16,f32)
        acc = __builtin_amdgcn_wmma_f32_16x16x4_f32(
            /*neg_a=*/false, a, /*neg_b=*/false, bb,
            /*c_mod=*/(short)0, acc, /*reuse_a=*/false, /*reuse_b=*/false);
    }

    // C/D layout: VGPR r, lanes 0-15 -> M=r, lanes 16-31 -> M=r+8; N = lane&15
#pragma unroll
    for (int r = 0; r < 8; ++r)
        lgt[wib][r + 8 * h][m] = acc[r];
    __syncthreads();

    if (lane < 16) {
        const int token = tokenBase + lane;    // token = b*N + n
        const int b = token >> 12;
        float lg[8];
#pragma unroll
        for (int e = 0; e < NE; ++e) lg[e] = lgt[wib][lane][e];

        float mx = lg[0];
#pragma unroll
        for (int e = 1; e < NE; ++e) mx = fmaxf(mx, lg[e]);
        float p[8], se = 0.f;
#pragma unroll
        for (int e = 0; e < NE; ++e) { p[e] = expf(lg[e] - mx); se += p[e]; }
        const float inv = 1.f / se;

        float g0 = -1.f; int i0 = 0;
#pragma unroll
        for (int e = 0; e < NE; ++e) {
            p[e] *= inv;
            if (p[e] > g0) { g0 = p[e]; i0 = e; }
        }
        float g1 = -1.f; int i1 = 0;
#pragma unroll
        for (int e = 0; e < NE; ++e)
            if (e != i0 && p[e] > g1) { g1 = p[e]; i1 = e; }

        const float denom = fmaxf(g0 + g1, 1e-9f);
        const float g0n = g0 / denom, g1n = g1 / denom;
        // should_route[1] = probs[1,b,n] < g1n / 0.2 ; slot 0 always routes
        const int route1 = (probs[NTOK + token] < g1n / 0.2f) ? 1 : 0;

        v4f r;
        r.x = g0n; r.y = g1n;
        r.z = __int_as_float(i0);
        r.w = __int_as_float(route1 ? i1 : -1);
        rec[token] = r;

        // loss accumulators
#pragma unroll
        for (int e = 0; e < NE; ++e) atomicAdd(&rawsum[b * NE + e], p[e]);
        const float lse = mx + logf(se);
        atomicAdd(zsum, lse * lse);
    }
}

// ---------------- Kernel 2: per-batch capacity scan + sparse scatter --------
// One block per batch; thread t owns tokens [t*16, t*16+16).
__global__ void k_scan(const v4f* __restrict__ rec, float* __restrict__ out,
                       float* __restrict__ cnt1) {
    __shared__ int lcnt[256][16];              // per-thread counts / excl prefixes
    __shared__ int totals[16];
    __shared__ int prevc[8];

    const int b = blockIdx.x;
    const int t = threadIdx.x;
    const int tok0 = b * NN + t * 16;

    // phase 1: local counts per (k,e) -> counter index p = k*8 + e
    int c[16];
#pragma unroll
    for (int p = 0; p < 16; ++p) c[p] = 0;
    for (int i = 0; i < 16; ++i) {
        v4f r = rec[tok0 + i];
        int i0 = __float_as_int(r.z);
        int i1 = __float_as_int(r.w);
        c[i0]++;
        if (i1 >= 0) c[8 + i1]++;
    }
#pragma unroll
    for (int p = 0; p < 16; ++p) lcnt[t][p] = c[p];
    __syncthreads();

    // phase 2: 16 serial exclusive scans over the 256 thread-chunks
    if (t < 16) {
        int run = 0;
        for (int tt = 0; tt < 256; ++tt) {
            int v = lcnt[tt][t];
            lcnt[tt][t] = run;
            run += v;
        }
        totals[t] = run;
    }
    __syncthreads();
    if (t < 8) {                                // prev_expert_count = clipped top-1 count
        int pc = totals[t] < CAP ? totals[t] : CAP;
        prevc[t] = pc;
        cnt1[b * NE + t] = (float)pc;           // density_1 numerator
    }
    __syncthreads();

    // phase 3: assign positions in sequence order, scatter nonzeros
    int run[16];
#pragma unroll
    for (int p = 0; p < 16; ++p) run[p] = lcnt[t][p];
    for (int i = 0; i < 16; ++i) {
        const int token = tok0 + i;             // == b*N + n
        v4f r = rec[token];
        const int i0 = __float_as_int(r.z);
        const int i1 = __float_as_int(r.w);

        int pos0 = run[i0]; run[i0]++;          // cumsum of raw mask (pre-clip)
        if (pos0 < CAP) {
            long off = ((long)token * NE + i0) * CAP + pos0;
            out[off]       = (r.x != 0.f) ? 1.f : 0.f;   // dispatch
            out[NDC + off] = r.x;                        // combine
        }
        if (i1 >= 0) {
            int pos1 = run[8 + i1] + prevc[i1]; run[8 + i1]++;
            if (pos1 < CAP) {
                long off = ((long)token * NE + i1) * CAP + pos1;
                out[off]       = (r.y != 0.f) ? 1.f : 0.f;
                out[NDC + off] = r.y;
            }
        }
    }
}

// ---------------- Kernel 3: scalar losses -----------------------------------
__global__ void k_final(const float* __restrict__ rawsum, const float* __restrict__ zsum,
                        const float* __restrict__ cnt1, float* __restrict__ out) {
    if (threadIdx.x == 0 && blockIdx.x == 0) {
        float bal = 0.f;
        for (int i = 0; i < NB * NE; ++i)
            bal += (rawsum[i] / (float)NN) * (cnt1[i] / (float)NN);
        out[2L * NDC]     = (bal / (float)(NB * NE)) * (float)(NE * NE); // balance_loss
        out[2L * NDC + 1] = zsum[0] / (float)NTOK;                       // router_z_loss
    }
}

// ---------------------------------------------------------------------------
extern "C" void kernel_launch(void* const* d_in, const int* in_sizes, int n_in,
                              void* d_out, int out_size, void* d_ws, size_t ws_size,
                              hipStream_t stream) {
    const float* x     = (const float*)d_in[0];   // [2,4096,2048]
    const float* W     = (const float*)d_in[1];   // [8,2048]
    const float* probs = (const float*)d_in[2];   // [2,2,4096]
    float* out = (float*)d_out;
    float* wsf = (float*)d_ws;

    v4f*   rec    = (v4f*)d_ws;                   // 8192 token records
    float* rawsum = wsf + WS_ACC_OFF;             // 16
    float* zsum   = wsf + WS_ACC_OFF + 16;        // 1
    float* cnt1   = wsf + WS_ACC_OFF + 32;        // 16

    k_zero <<<8192, 256, 0, stream>>>(out, rawsum);
    k_gate <<<64,   256, 0, stream>>>(x, W, probs, rec, rawsum, zsum);
    k_scan <<<NB,   256, 0, stream>>>(rec, out, cnt1);
    k_final<<<1,    32,  0, stream>>>(rawsum, zsum, cnt1, out);
}